// SphereValuedVolterra_40372692583158
// MI455X (gfx1250) — compile-verified
//
#include <hip/hip_runtime.h>

typedef float v2f __attribute__((ext_vector_type(2)));
typedef float v8f __attribute__((ext_vector_type(8)));

#define IN_CH 16
#define OUT_CH 32
#define KW 3
#define NN 144          // KW*KW*IN_CH
#define HP 62
#define WP 62
#define PT 16           // output columns per block tile
#define EPSF 1e-7f

__global__ __launch_bounds__(256) void volterra_sphere_kernel(
    const float* __restrict__ x,    // (2, 16, 64, 64, 3) f32, unit vectors
    const float* __restrict__ W,    // (3, 32, 144) f32
    float* __restrict__ out)        // (2, 32, 62, 62, 3) f32
{
    const int wtile = blockIdx.x;       // 0..3
    const int h     = blockIdx.y;       // 0..61
    const int b     = blockIdx.z;       // 0..1
    const int w0c   = wtile * PT;
    const int t     = threadIdx.x;      // 0..255 (8 waves of 32)

    __shared__ float sW0[OUT_CH][NN + 1];          // relu+normalized w0; 145 stride -> no bank conflicts
    __shared__ float sX[KW][IN_CH][PT + 2][3];     // input window rows h..h+2, cols w0c..w0c+17
    __shared__ float sM[OUT_CH][PT][3];            // raw anchors from WMMA
    __shared__ float sInv[OUT_CH];

    // ---- stage relu(W0) and the input window into LDS ----
    for (int i = t; i < OUT_CH * NN; i += 256) {
        int o = i / NN, n = i - o * NN;
        sW0[o][n] = fmaxf(W[i], 0.0f);
    }
    for (int i = t; i < KW * IN_CH * (PT + 2) * 3; i += 256) {
        int e = i;
        int d   = e % 3;        e /= 3;
        int col = e % (PT + 2); e /= (PT + 2);
        int c   = e % IN_CH;    e /= IN_CH;
        int ki  = e;
        int gcol = w0c + col;
        float v = 0.0f;
        if (gcol < 64)
            v = x[((((b * IN_CH + c) * 64) + (h + ki)) * 64 + gcol) * 3 + d];
        sX[ki][c][col][d] = v;
    }
    __syncthreads();

    // ---- normalize w0 rows to convex weights:  w0 / (sum + eps) ----
    if (t < OUT_CH) {
        float s = 0.0f;
        for (int n = 0; n < NN; ++n) s += sW0[t][n];
        sInv[t] = 1.0f / (s + EPSF);
    }
    __syncthreads();
    for (int i = t; i < OUT_CH * NN; i += 256) {
        int o = i / NN, n = i - o * NN;
        sW0[o][n] *= sInv[o];
    }
    __syncthreads();

    // ---- anchors: m_raw[o][p][d] = sum_n w0[o][n] * xw[p][n][d] ----
    // Six 16x16 (channels-half x positions) tiles, K=144, via v_wmma_f32_16x16x4_f32.
    {
        const int lane = t & 31;
        const int wv   = t >> 5;
        if (wv < 6) {                       // wave-uniform branch: EXEC all ones inside
            const int hf  = wv & 1;         // channel half (0: o=0..15, 1: o=16..31)
            const int dd  = wv >> 1;        // vector component 0..2
            const int row = lane & 15;      // A: M index; B/D: N index (position)
            const int kg  = lane >> 4;      // K sub-group
            const int o   = row + 16 * hf;
            v8f acc = {0.f, 0.f, 0.f, 0.f, 0.f, 0.f, 0.f, 0.f};
            for (int nb = 0; nb < NN / 4; ++nb) {
                const int n0 = nb * 4 + 2 * kg;
                v2f a, bm;
                // A frag: lane holds M=row, K = 2*kg + r
                a.x = sW0[o][n0];
                a.y = sW0[o][n0 + 1];
                // B frag: lane holds K = 2*kg + r, N = row (= position p)
#pragma unroll
                for (int r = 0; r < 2; ++r) {
                    int n  = n0 + r;
                    int kk = n >> 4;        // kernel position 0..8
                    int c  = n & 15;        // channel-minor (matches reference unfold)
                    int ki = kk / 3;
                    int kj = kk - ki * 3;
                    float val = sX[ki][c][row + kj][dd];
                    if (r == 0) bm.x = val; else bm.y = val;
                }
                acc = __builtin_amdgcn_wmma_f32_16x16x4_f32(
                    false, a, false, bm, (short)0, acc, false, false);
            }
            // D layout: VGPR i -> M = i + 8*kg, N = row
#pragma unroll
            for (int i = 0; i < 8; ++i) {
                int oo = i + 8 * kg + 16 * hf;
                sM[oo][row][dd] = acc[i];
            }
        }
    }
    __syncthreads();

    // ---- per (channel, position): log-map, Volterra combine, exp-map ----
    const int o = t & 31;                   // lanes of a wave share p -> LDS broadcasts
    const float* __restrict__ w1p = W + (OUT_CH + o) * NN;
    const float* __restrict__ w2p = W + (2 * OUT_CH + o) * NN;

    for (int pb = 0; pb < PT; pb += 8) {
        const int p  = (t >> 5) + pb;
        const int wg = w0c + p;
        if (wg >= WP) continue;

        float mx = sM[o][p][0], my = sM[o][p][1], mz = sM[o][p][2];
        float inv = 1.0f / (sqrtf(mx * mx + my * my + mz * mz) + EPSF);
        mx *= inv; my *= inv; mz *= inv;

        float l0x = 0.f, l0y = 0.f, l0z = 0.f, s0 = 0.f;
        float l1x = 0.f, l1y = 0.f, l1z = 0.f, s1 = 0.f;
        float l2x = 0.f, l2y = 0.f, l2z = 0.f, s2 = 0.f;

        int n = 0;
        for (int ki = 0; ki < KW; ++ki)
            for (int kj = 0; kj < KW; ++kj) {
                const float* xr = &sX[ki][0][p + kj][0];    // stride per channel: (PT+2)*3
                for (int c = 0; c < IN_CH; ++c, ++n) {
                    float xx = xr[c * ((PT + 2) * 3) + 0];
                    float xy = xr[c * ((PT + 2) * 3) + 1];
                    float xz = xr[c * ((PT + 2) * 3) + 2];
                    float ct = mx * xx + my * xy + mz * xz;
                    ct = fminf(fmaxf(ct, -1.0f + 1e-6f), 1.0f - 1e-6f);
                    float th = acosf(ct);
                    float g  = th / sinf(th);               // log-map scale
                    float a0 = sW0[o][n] * g;
                    float a1 = fmaxf(w1p[n], 0.0f) * g;
                    float a2 = fmaxf(w2p[n], 0.0f) * g;
                    l0x = fmaf(a0, xx, l0x); l0y = fmaf(a0, xy, l0y); l0z = fmaf(a0, xz, l0z);
                    s0  = fmaf(a0, ct, s0);
                    l1x = fmaf(a1, xx, l1x); l1y = fmaf(a1, xy, l1y); l1z = fmaf(a1, xz, l1z);
                    s1  = fmaf(a1, ct, s1);
                    l2x = fmaf(a2, xx, l2x); l2y = fmaf(a2, xy, l2y); l2z = fmaf(a2, xz, l2z);
                    s2  = fmaf(a2, ct, s2);
                }
            }

        float u0x = l0x - s0 * mx, u0y = l0y - s0 * my, u0z = l0z - s0 * mz;
        float u1x = l1x - s1 * mx, u1y = l1y - s1 * my, u1z = l1z - s1 * mz;
        float u2x = l2x - s2 * mx, u2y = l2y - s2 * my, u2z = l2z - s2 * mz;
        float vx = u0x + u1x * u2x, vy = u0y + u1y * u2y, vz = u0z + u1z * u2z;
        float dm = vx * mx + vy * my + vz * mz;             // re-project onto T_m S^2
        vx -= dm * mx; vy -= dm * my; vz -= dm * mz;
        float nv = sqrtf(vx * vx + vy * vy + vz * vz);
        float sc = sinf(nv) / (nv + EPSF);
        float cn = cosf(nv);
        float yx = cn * mx + sc * vx;
        float yy = cn * my + sc * vy;
        float yz = cn * mz + sc * vz;

        size_t oi = ((((size_t)b * OUT_CH + o) * HP + h) * WP + wg) * 3;
        out[oi + 0] = yx;
        out[oi + 1] = yy;
        out[oi + 2] = yz;
    }
}

extern "C" void kernel_launch(void* const* d_in, const int* in_sizes, int n_in,
                              void* d_out, int out_size, void* d_ws, size_t ws_size,
                              hipStream_t stream) {
    const float* x = (const float*)d_in[0];     // (2,16,64,64,3) f32
    const float* W = (const float*)d_in[1];     // (3,32,144) f32
    float* out = (float*)d_out;                 // (2,32,62,62,3) f32
    dim3 grid((WP + PT - 1) / PT, HP, 2);       // 4 x 62 x 2 blocks
    volterra_sphere_kernel<<<grid, 256, 0, stream>>>(x, W, out);
    (void)in_sizes; (void)n_in; (void)out_size; (void)d_ws; (void)ws_size;
}